// PyTorchSineGen_65936337928793
// MI455X (gfx1250) — compile-verified
//
#include <hip/hip_runtime.h>
#include <cstdint>

// ---------------- problem constants ----------------
constexpr int   B_      = 16;
constexpr int   L_      = 480000;
constexpr int   DIM_    = 9;          // harmonics 1..9
constexpr float INV_SR  = 1.0f / 24000.0f;
constexpr float SINE_AMP = 0.1f;

constexpr int   THREADS = 256;        // 8 wave32
constexpr int   ITER    = 32;         // elements per thread per chunk
constexpr int   CHUNK   = THREADS * ITER;              // 8192
constexpr int   NCHUNK  = (L_ + CHUNK - 1) / CHUNK;    // 59 (tail = 4864 = 19*256)

typedef float v4f __attribute__((ext_vector_type(4)));

// ---------------------------------------------------------------------------
// Pass 1: per-(batch,chunk) partial sums of f0/SR.
// Full chunks: fixed-trip unrolled float4 loads (8 independent b128 in flight).
// ---------------------------------------------------------------------------
__global__ __launch_bounds__(THREADS)
void sine_pass1_chunksum(const float* __restrict__ f0, float* __restrict__ chunkSums) {
    const int blk  = blockIdx.x;
    const int b    = blk / NCHUNK;
    const int c    = blk % NCHUNK;
    const int base = c * CHUNK;
    const int n    = min(CHUNK, L_ - base);
    const float* src = f0 + (size_t)b * L_ + base;
    const int tid  = threadIdx.x;

    float acc = 0.0f;
    if (n == CHUNK) {
        const v4f* s4 = (const v4f*)src;           // 16B-aligned (base % 8192 == 0)
        #pragma unroll
        for (int k = 0; k < ITER / 4; ++k) {
            v4f v = s4[k * THREADS + tid];
            acc += v.x + v.y + v.z + v.w;
        }
    } else {
        for (int i = tid; i < n; i += THREADS) acc += src[i];
    }

    // wave32 reduction
    #pragma unroll
    for (int off = 16; off > 0; off >>= 1) acc += __shfl_xor(acc, off, 32);

    __shared__ float wsum[THREADS / 32];
    const int wave = tid >> 5;
    const int lane = tid & 31;
    if (lane == 0) wsum[wave] = acc;
    __syncthreads();
    if (tid == 0) {
        float t = 0.0f;
        #pragma unroll
        for (int w = 0; w < THREADS / 32; ++w) t += wsum[w];
        chunkSums[blk] = t * INV_SR;
    }
}

// ---------------------------------------------------------------------------
// Pass 2: per-batch exclusive scan of chunk sums (double accum, wrapped to [0,1))
// ---------------------------------------------------------------------------
__global__ __launch_bounds__(64)
void sine_pass2_scan(const float* __restrict__ chunkSums, float* __restrict__ chunkPrefix) {
    const int b = blockIdx.x;
    __shared__ float s[NCHUNK];
    __shared__ float p[NCHUNK];
    for (int i = threadIdx.x; i < NCHUNK; i += 64) s[i] = chunkSums[b * NCHUNK + i];
    __syncthreads();
    if (threadIdx.x == 0) {
        double acc = 0.0;
        for (int i = 0; i < NCHUNK; ++i) {
            p[i] = (float)(acc - floor(acc));
            acc += (double)s[i];
        }
    }
    __syncthreads();
    for (int i = threadIdx.x; i < NCHUNK; i += 64) chunkPrefix[b * NCHUNK + i] = p[i];
}

// ---------------------------------------------------------------------------
// Pass 3: async-stage f0 tile to LDS (ASYNCcnt), block scan, 9x v_sin_f32,
// LDS-transposed fully-coalesced 128-bit non-temporal streaming stores.
// ---------------------------------------------------------------------------
__global__ __launch_bounds__(THREADS)
void sine_pass3_generate(const float* __restrict__ f0,
                         const float* __restrict__ rand_ini,
                         const float* __restrict__ chunkPrefix,
                         float* __restrict__ outSine,
                         float* __restrict__ outUv) {
    const int blk  = blockIdx.x;
    const int b    = blk / NCHUNK;
    const int c    = blk % NCHUNK;
    const int base = c * CHUNK;
    const int n    = min(CHUNK, L_ - base);   // 8192 or 4864 -- both multiples of 256
    const int tid  = threadIdx.x;
    const int wave = tid >> 5;
    const int lane = tid & 31;

    __shared__ float tile[CHUNK];                    // 32 KB f0 staging
    __shared__ float obuf[2][THREADS * DIM_];        // 2 x 9 KB output staging
    __shared__ float wsum[THREADS / 32];

    // ---- CDNA5 async global->LDS copy of the f0 tile (tracked by ASYNCcnt) ----
    {
        const unsigned ldsBase = (unsigned)(uintptr_t)&tile[0];
        const float* src = f0 + (size_t)b * L_ + base;
        #pragma unroll
        for (int k = 0; k < ITER / 4; ++k) {         // up to 8 x b128 per lane
            const int e = k * THREADS * 4 + tid * 4;
            if (e < n) {
                const unsigned lds = ldsBase + (unsigned)e * 4u;
                const unsigned long long ga = (unsigned long long)(uintptr_t)(src + e);
                asm volatile("global_load_async_to_lds_b128 %0, %1, off"
                             :: "v"(lds), "v"(ga) : "memory");
            }
        }
        asm volatile("s_wait_asynccnt 0" ::: "memory");
    }
    __syncthreads();   // tile valid for all waves

    // per-batch harmonic initial phases
    float rini[DIM_];
    #pragma unroll
    for (int d = 0; d < DIM_; ++d) rini[d] = rand_ini[b * DIM_ + d];

    // running carry (uniform over block), starts at inter-chunk prefix in [0,1)
    float carry = chunkPrefix[b * NCHUNK + c];

    const int nIter = n / THREADS;   // 32 or 19 -- iterations are uniformly valid
    for (int i = 0; i < nIter; ++i) {
        const int idx = i * THREADS + tid;
        const float f0v = tile[idx];
        float v = f0v * INV_SR;

        // inclusive wave32 scan
        #pragma unroll
        for (int off = 1; off < 32; off <<= 1) {
            const float y = __shfl_up(v, off, 32);
            if (lane >= off) v += y;
        }
        if (lane == 31) wsum[wave] = v;
        __syncthreads();                                   // (A) wsum ready

        float pre = carry;
        for (int w = 0; w < wave; ++w) pre += wsum[w];
        float total = 0.0f;
        #pragma unroll
        for (int w = 0; w < THREADS / 32; ++w) total += wsum[w];
        carry += total;
        carry -= floorf(carry);                            // keep carry wrapped

        const float s   = pre + v;                         // inclusive prefix
        const float sfr = s - floorf(s);                   // phase base in turns
        const float uvf = (f0v > 0.0f) ? 1.0f : 0.0f;
        const float amp = SINE_AMP * uvf;

        float* ob = &obuf[i & 1][tid * DIM_];              // stride 9: bank-conflict-free
        #pragma unroll
        for (int d = 0; d < DIM_; ++d) {
            float t = sfr * (float)(d + 1) + rini[d];
            t -= floorf(t);                                // [0,1) turns
            ob[d] = amp * __builtin_amdgcn_sinf(t);        // v_sin_f32 = sin(2*pi*x)
        }
        __builtin_nontemporal_store(uvf, outUv + (size_t)b * L_ + base + idx);
        __syncthreads();                                   // (C) obuf ready, wsum reads done

        // fully-coalesced 128-bit NT stores of the contiguous 9216B region
        // region byte offset = 36*(b*L + base + i*256), provably 16B-aligned
        v4f* dst4 = (v4f*)(outSine + ((size_t)b * L_ + base + (size_t)i * THREADS) * DIM_);
        const v4f* sb4 = (const v4f*)&obuf[i & 1][0];
        for (int j = tid; j < (THREADS * DIM_) / 4; j += THREADS) {  // 576 v4f
            __builtin_nontemporal_store(sb4[j], dst4 + j);
        }
        // no barrier needed here: obuf double-buffered, next wsum write fenced by (C)
    }
}

// ---------------------------------------------------------------------------
extern "C" void kernel_launch(void* const* d_in, const int* in_sizes, int n_in,
                              void* d_out, int out_size, void* d_ws, size_t ws_size,
                              hipStream_t stream) {
    const float* f0       = (const float*)d_in[0];   // [B, L]
    const float* rand_ini = (const float*)d_in[1];   // [B, 1, DIM]
    float* outSine = (float*)d_out;                              // [B, L, DIM]
    float* outUv   = outSine + (size_t)B_ * L_ * DIM_;           // [B, L, 1]

    float* chunkSums   = (float*)d_ws;               // B*NCHUNK floats
    float* chunkPrefix = chunkSums + B_ * NCHUNK;    // B*NCHUNK floats

    const int nblk = B_ * NCHUNK;
    sine_pass1_chunksum<<<nblk, THREADS, 0, stream>>>(f0, chunkSums);
    sine_pass2_scan<<<B_, 64, 0, stream>>>(chunkSums, chunkPrefix);
    sine_pass3_generate<<<nblk, THREADS, 0, stream>>>(f0, rand_ini, chunkPrefix,
                                                      outSine, outUv);
}